// PositionalAttentionHead_9766755631240
// MI455X (gfx1250) — compile-verified
//
#include <hip/hip_runtime.h>

#define BB 4
#define SS 1024
#define NHH 16
#define HDD 64
#define HIDD 1024

typedef float v2f __attribute__((ext_vector_type(2)));
typedef float v4f __attribute__((ext_vector_type(4)));
typedef float v8f __attribute__((ext_vector_type(8)));

static __device__ __forceinline__ v8f wmma4(v2f a, v2f b, v8f c) {
  // D = A(16x4 f32) * B(4x16 f32) + C(16x16 f32)
  return __builtin_amdgcn_wmma_f32_16x16x4_f32(false, a, false, b, (short)0, c,
                                               false, false);
}

// ---------------------------------------------------------------------------
// Kernel 1: per-token LN over head_dim, QKV projection via WMMA, RoPE on Q/K.
// One wave (32 threads) per (b, s); its 16 rows are the 16 heads of the token,
// which are the contiguous 1024 floats x[tile*1024 .. +1024).
// Writes Q,K as [b,h,s,d] and V transposed as [b,h,d,s].
// ---------------------------------------------------------------------------
__global__ __launch_bounds__(32) void qkv_rope_kernel(
    const float* __restrict__ x, const float* __restrict__ gamma,
    const float* __restrict__ beta, const float* __restrict__ Wq,
    const float* __restrict__ Wk, const float* __restrict__ Wv,
    float* __restrict__ Qo, float* __restrict__ Ko, float* __restrict__ Vt) {
  __shared__ __align__(16) float xn[16 * 68];

  const int l = threadIdx.x;        // 0..31
  const int tile = blockIdx.x;      // b*1024 + s
  const int b = tile >> 10;
  const int spos = tile & 1023;
  const int half = l & 1;           // which 32-dim half of the head row
  const int hrow = l >> 1;          // head row 0..15 (LN phase)

  // ---- LayerNorm over head_dim=64 (2 lanes per row, 32 elems each) ----
  // x is read exactly once across the whole dispatch -> non-temporal loads
  // so it does not displace W/gamma/beta (and later Q/K/Vt) in cache.
  const float* xrow = x + (size_t)tile * 1024 + (size_t)l * 32;
  v4f xq[8];
#pragma unroll
  for (int j = 0; j < 8; ++j)
    xq[j] = __builtin_nontemporal_load((const v4f*)(xrow + 4 * j));

  float sm = 0.f;
#pragma unroll
  for (int j = 0; j < 8; ++j) sm += xq[j].x + xq[j].y + xq[j].z + xq[j].w;
  sm += __shfl_xor(sm, 1, 32);
  const float mean = sm * (1.0f / 64.0f);

  float vr = 0.f;
#pragma unroll
  for (int j = 0; j < 8; ++j) {
    float d0 = xq[j].x - mean, d1 = xq[j].y - mean;
    float d2 = xq[j].z - mean, d3 = xq[j].w - mean;
    vr += d0 * d0 + d1 * d1 + d2 * d2 + d3 * d3;
  }
  vr += __shfl_xor(vr, 1, 32);
  const float rs = rsqrtf(vr * (1.0f / 64.0f) + 1e-5f);

#pragma unroll
  for (int j = 0; j < 8; ++j) {
    v4f g = *(const v4f*)(gamma + half * 32 + 4 * j);
    v4f be = *(const v4f*)(beta + half * 32 + 4 * j);
    v4f o;
    o.x = (xq[j].x - mean) * rs * g.x + be.x;
    o.y = (xq[j].y - mean) * rs * g.y + be.y;
    o.z = (xq[j].z - mean) * rs * g.z + be.z;
    o.w = (xq[j].w - mean) * rs * g.w + be.w;
    *(v4f*)(xn + hrow * 68 + half * 32 + 4 * j) = o;
  }
  __syncthreads();

  // ---- WMMA fragment addressing ----
  const int m = l & 15;          // A/B row index (head for A, out-dim for B)
  const int kh = (l >> 4) * 2;   // K sub-offset: lanes 0-15 -> 0, 16-31 -> 2
  const int hb = (l >> 4) * 8;   // D: M = v + hb

  // A fragments: xn tile, A(m = head, k = in-dim)
  v2f a[16];
#pragma unroll
  for (int kk = 0; kk < 16; ++kk)
    a[kk] = *(const v2f*)(xn + m * 68 + kk * 4 + kh);

  v8f acc[4];
  const v8f zero = {0.f, 0.f, 0.f, 0.f, 0.f, 0.f, 0.f, 0.f};

  // helper: acc[nt] = xn(16x64) @ W^T tile nt; B(k,n) = W[n][k]
  auto proj = [&](const float* __restrict__ W) {
#pragma unroll
    for (int nt = 0; nt < 4; ++nt) {
      v8f c = zero;
      const float* wb = W + (nt * 16 + m) * 64 + kh;
#pragma unroll
      for (int kk = 0; kk < 16; ++kk) {
        v2f bf = *(const v2f*)(wb + kk * 4);
        c = wmma4(a[kk], bf, c);
      }
      acc[nt] = c;
    }
  };

  // ---- V: no RoPE, store transposed Vt[b][h][d][s] ----
  proj(Wv);
#pragma unroll
  for (int nt = 0; nt < 4; ++nt)
#pragma unroll
    for (int v = 0; v < 8; ++v)
      Vt[((size_t)((b * 16 + (v + hb)) * 64 + (m + 16 * nt))) * 1024 + spos] =
          acc[nt][v];

  // ---- RoPE angles (shared by Q and K) ----
  const float LOG1E4_32 = 0.28782313662425575f;  // ln(10000)/32
  const float sf = (float)spos;
  const float ang0 = sf * expf(-(float)m * LOG1E4_32);          // e%32 = m
  const float ang1 = sf * expf(-(float)(m + 16) * LOG1E4_32);   // e%32 = m+16
  const float c0 = cosf(ang0), s0 = sinf(ang0);
  const float c1 = cosf(ang1), s1 = sinf(ang1);

  auto rope = [&]() {
#pragma unroll
    for (int v = 0; v < 8; ++v) {
      float lo = acc[0][v], hi = acc[2][v];
      acc[0][v] = lo * c0 - hi * s0;
      acc[2][v] = hi * c0 + lo * s0;
      lo = acc[1][v]; hi = acc[3][v];
      acc[1][v] = lo * c1 - hi * s1;
      acc[3][v] = hi * c1 + lo * s1;
    }
  };

  // ---- Q ----
  proj(Wq);
  rope();
#pragma unroll
  for (int nt = 0; nt < 4; ++nt)
#pragma unroll
    for (int v = 0; v < 8; ++v)
      Qo[((size_t)((b * 16 + (v + hb)) * 1024 + spos)) * 64 + m + 16 * nt] =
          acc[nt][v];

  // ---- K ----
  proj(Wk);
  rope();
#pragma unroll
  for (int nt = 0; nt < 4; ++nt)
#pragma unroll
    for (int v = 0; v < 8; ++v)
      Ko[((size_t)((b * 16 + (v + hb)) * 1024 + spos)) * 64 + m + 16 * nt] =
          acc[nt][v];
}

// ---------------------------------------------------------------------------
// Kernel 2: attention for one (b, h, 16-query tile). 128 threads = 4 waves.
// LDS holds the 16x1024 score row (stride 1028 for bank-conflict-free WMMA
// stores and PV A-fragment reads). Softmax matches reference:
// exp(s - max) / (sum + 1e-7). Probs streamed to global coalesced with
// non-temporal stores (write-once 268MB stream must not evict Q/K/Vt in L2).
// ---------------------------------------------------------------------------
__global__ __launch_bounds__(128) void attn_kernel(
    const float* __restrict__ Qo, const float* __restrict__ Ko,
    const float* __restrict__ Vt, const unsigned char* __restrict__ mask,
    float* __restrict__ out, float* __restrict__ probs) {
  extern __shared__ float sc[];  // 16 * 1028 floats

  const int bid = blockIdx.x;
  const int b = bid >> 10;
  const int h = (bid >> 6) & 15;
  const int q0 = (bid & 63) * 16;

  const int tid = threadIdx.x;
  const int w = tid >> 5;       // wave 0..3
  const int l = tid & 31;
  const int m = l & 15;
  const int kh = (l >> 4) * 2;
  const int hb = (l >> 4) * 8;

  const float* qbase = Qo + ((size_t)(b * 16 + h) * 1024 + q0) * 64;
  const float* kbase = Ko + ((size_t)(b * 16 + h) * 1024) * 64;
  const float* vtb = Vt + ((size_t)(b * 16 + h) * 64) * 1024;
  const unsigned char* mrow = mask + b * 1024;

  const v8f zero = {0.f, 0.f, 0.f, 0.f, 0.f, 0.f, 0.f, 0.f};
  const float NEG_INF = -__builtin_inff();

  // Q fragments for this query tile (A: m = query row, k = dim)
  v2f qa[16];
#pragma unroll
  for (int kk = 0; kk < 16; ++kk)
    qa[kk] = *(const v2f*)(qbase + m * 64 + kk * 4 + kh);

  // ---- scores = Q K^T / 8, masked; wave w handles key tiles w, w+4, ... ----
  for (int j = w; j < 64; j += 4) {
    v8f acc = zero;
    const float* kb = kbase + (j * 16 + m) * 64 + kh;  // B(k,n)=K[key=n][k]
#pragma unroll
    for (int kk = 0; kk < 16; ++kk) {
      v2f kf = *(const v2f*)(kb + kk * 4);
      acc = wmma4(qa[kk], kf, acc);
    }
    const int col = j * 16 + m;                 // key index (per lane)
    const bool keep = mrow[col] != 0;
#pragma unroll
    for (int v = 0; v < 8; ++v) {
      float sval = acc[v] * 0.125f;
      if (!keep) sval = NEG_INF;
      sc[(v + hb) * 1028 + col] = sval;         // row = query v+hb, col = key
    }
  }
  __syncthreads();

  // ---- softmax: wave w owns rows 4w..4w+3; 8 lanes per row ----
  const int rloc = l >> 3;       // 0..3
  const int sub = l & 7;         // 0..7 segment within row
  float* rowp = sc + (w * 4 + rloc) * 1028;

  float mx = NEG_INF;
#pragma unroll 4
  for (int i = 0; i < 128; ++i) mx = fmaxf(mx, rowp[sub + 8 * i]);
  mx = fmaxf(mx, __shfl_xor(mx, 1, 32));
  mx = fmaxf(mx, __shfl_xor(mx, 2, 32));
  mx = fmaxf(mx, __shfl_xor(mx, 4, 32));

  float smv = 0.f;
#pragma unroll 4
  for (int i = 0; i < 128; ++i) {
    float e = __expf(rowp[sub + 8 * i] - mx);
    rowp[sub + 8 * i] = e;
    smv += e;
  }
  smv += __shfl_xor(smv, 1, 32);
  smv += __shfl_xor(smv, 2, 32);
  smv += __shfl_xor(smv, 4, 32);
  const float inv = 1.0f / (smv + 1e-7f);

  // ---- normalize + coalesced non-temporal global write of probs ----
  float* pbase = probs + ((size_t)(b * 16 + h) * 1024 + q0) * 1024;
#pragma unroll
  for (int rr = 0; rr < 4; ++rr) {
    const int rq = w * 4 + rr;
    const float invr = __shfl(inv, rr * 8, 32);
    float* rp = sc + rq * 1028;
    float* gp = pbase + (size_t)rq * 1024;
#pragma unroll 4
    for (int i = 0; i < 32; ++i) {
      const int c = l + 32 * i;
      const float p = rp[c] * invr;
      rp[c] = p;
      __builtin_nontemporal_store(p, gp + c);
    }
  }
  __syncthreads();

  // ---- out tile = probs(16x1024) @ V(1024x64); wave w owns dims 16w..16w+15
  v8f oacc = zero;
  const float* vrow = vtb + (w * 16 + m) * 1024 + kh;  // B(k,n)=V[k][n]=Vt[n][k]
  for (int kc = 0; kc < 256; ++kc) {
    v2f pa = *(const v2f*)(sc + m * 1028 + kc * 4 + kh);
    v2f vb = *(const v2f*)(vrow + kc * 4);
    oacc = wmma4(pa, vb, oacc);
  }
#pragma unroll
  for (int v = 0; v < 8; ++v)
    __builtin_nontemporal_store(
        oacc[v],
        out + ((size_t)(b * 1024 + q0 + v + hb)) * 1024 + h * 64 + w * 16 + m);
}

// ---------------------------------------------------------------------------
extern "C" void kernel_launch(void* const* d_in, const int* in_sizes, int n_in,
                              void* d_out, int out_size, void* d_ws,
                              size_t ws_size, hipStream_t stream) {
  (void)in_sizes; (void)n_in; (void)out_size; (void)ws_size;

  const float* x = (const float*)d_in[0];
  const unsigned char* mask = (const unsigned char*)d_in[1];  // bool[B,S]
  const float* gamma = (const float*)d_in[2];
  const float* beta = (const float*)d_in[3];
  const float* Wq = (const float*)d_in[4];
  const float* Wk = (const float*)d_in[5];
  const float* Wv = (const float*)d_in[6];

  float* out = (float*)d_out;                       // [B,S,HID]
  float* probs = out + (size_t)BB * SS * HIDD;      // [B,NH,S,S]

  float* qb = (float*)d_ws;                         // [B,NH,S,HD]
  float* kb = qb + (size_t)BB * NHH * SS * HDD;     // [B,NH,S,HD]
  float* vt = kb + (size_t)BB * NHH * SS * HDD;     // [B,NH,HD,S] (transposed)

  qkv_rope_kernel<<<BB * SS, 32, 0, stream>>>(x, gamma, beta, Wq, Wk, Wv, qb,
                                              kb, vt);

  const int blocks = BB * NHH * (SS / 16);          // 4096
  const size_t lds = (size_t)16 * 1028 * sizeof(float);  // 65792 B
  attn_kernel<<<blocks, 128, lds, stream>>>(qb, kb, vt, mask, out, probs);
}